// TripletLoss_37873021616787
// MI455X (gfx1250) — compile-verified
//
#include <hip/hip_runtime.h>

// Clang ext vectors (HIP's float4 is a class; __builtin_nontemporal_load needs a
// real vector type, and the WMMA builtin needs exact ext_vector types).
typedef float v2f __attribute__((ext_vector_type(2)));
typedef float v4f __attribute__((ext_vector_type(4)));
typedef float v8f __attribute__((ext_vector_type(8)));

#define EMBED_D 4096
#define THREADS 256
#define TRIPLET_MARGIN 1.0f

// ---------------------------------------------------------------------------
// Kernel 1: one 256-thread block (8 waves, wave32) per triplet.
// Streams 48KB of contiguous memory (rows 3t,3t+1,3t+2) with NT b128 loads,
// accumulates squared distances, reduces, writes per-triplet hinge to ws.
// Bandwidth-bound: 402.7 MB @ 23.3 TB/s => ~17us floor.
// ---------------------------------------------------------------------------
__global__ __launch_bounds__(THREADS)
void triplet_hinge_kernel(const float* __restrict__ x, float* __restrict__ hinge) {
    const int t = blockIdx.x;                       // triplet id
    const size_t base = (size_t)t * 3u * EMBED_D;
    const v4f* __restrict__ q = (const v4f*)(x + base);
    const v4f* __restrict__ p = (const v4f*)(x + base + EMBED_D);
    const v4f* __restrict__ n = (const v4f*)(x + base + 2 * EMBED_D);

    const int tid = threadIdx.x;
    float sqp = 0.0f, sqn = 0.0f;

#pragma unroll
    for (int j = 0; j < (EMBED_D / 4) / THREADS; ++j) {   // 4 iterations
        const int i = tid + j * THREADS;
        v4f qv = __builtin_nontemporal_load(q + i);
        v4f pv = __builtin_nontemporal_load(p + i);
        v4f nv = __builtin_nontemporal_load(n + i);
        v4f dp = qv - pv;
        v4f dn = qv - nv;
        sqp = fmaf(dp.x, dp.x, fmaf(dp.y, dp.y, fmaf(dp.z, dp.z, fmaf(dp.w, dp.w, sqp))));
        sqn = fmaf(dn.x, dn.x, fmaf(dn.y, dn.y, fmaf(dn.z, dn.z, fmaf(dn.w, dn.w, sqn))));
    }

    // wave32 reduction (warpSize == 32 on gfx1250)
#pragma unroll
    for (int off = 16; off > 0; off >>= 1) {
        sqp += __shfl_down(sqp, off, 32);
        sqn += __shfl_down(sqn, off, 32);
    }

    __shared__ float sA[THREADS / 32];
    __shared__ float sB[THREADS / 32];
    const int wave = tid >> 5;
    const int lane = tid & 31;
    if (lane == 0) { sA[wave] = sqp; sB[wave] = sqn; }
    __syncthreads();

    if (tid == 0) {
        float a = 0.0f, b = 0.0f;
#pragma unroll
        for (int w = 0; w < THREADS / 32; ++w) { a += sA[w]; b += sB[w]; }
        hinge[t] = fmaxf(TRIPLET_MARGIN + sqrtf(a) - sqrtf(b), 0.0f);
    }
}

// ---------------------------------------------------------------------------
// Kernel 2: deterministic 8192 -> 1 reduction via V_WMMA_F32_16X16X4_F32.
// A(16x4) = 64 data values/iter (any slot permutation is fine for a sum),
// B(4x16) = ones  =>  D[m][n] = sum_k A[m][k], accumulated into C.
// C/D layout: lanes 0-15 hold rows 0-7 in VGPRs 0-7, lanes 16-31 hold rows
// 8-15, so total = (sum of 8 accs) + shfl_xor(.,16). Single wave, EXEC all-1s.
// ---------------------------------------------------------------------------
__global__ __launch_bounds__(32)
void final_reduce_wmma(const float* __restrict__ v, float* __restrict__ out, int n) {
    const int lane = threadIdx.x;
    v2f ones; ones.x = 1.0f; ones.y = 1.0f;
    v8f c = {0.f, 0.f, 0.f, 0.f, 0.f, 0.f, 0.f, 0.f};

    const int chunks = n >> 6;              // 64 values per WMMA
    const v2f* v2 = (const v2f*)v;
    for (int i = 0; i < chunks; ++i) {
        v2f a = v2[i * 32 + lane];
        // (neg_a, A, neg_b, B, c_mod, C, reuse_a, reuse_b)
        c = __builtin_amdgcn_wmma_f32_16x16x4_f32(
            false, a, false, ones, (short)0, c, false, false);
    }

    float s = c[0] + c[1] + c[2] + c[3] + c[4] + c[5] + c[6] + c[7];
    s += __shfl_xor(s, 16, 32);             // fold rows 0-7 with rows 8-15

    if (lane == 0) {
        for (int i = chunks << 6; i < n; ++i) s += v[i];   // tail (none for 8192)
        out[0] = fmaxf(s / (float)n, 0.0f);
    }
}

// ---------------------------------------------------------------------------
extern "C" void kernel_launch(void* const* d_in, const int* in_sizes, int n_in,
                              void* d_out, int out_size, void* d_ws, size_t ws_size,
                              hipStream_t stream) {
    (void)n_in; (void)out_size; (void)ws_size;
    const float* x = (const float*)d_in[0];    // f32[24576, 4096]; d_in[1] (y) unused
    float* out = (float*)d_out;                // f32[1]
    float* hinge = (float*)d_ws;               // 8192 floats of scratch

    const int B = in_sizes[0] / EMBED_D;       // 24576
    const int T = B / 3;                       // 8192 triplets

    triplet_hinge_kernel<<<T, THREADS, 0, stream>>>(x, hinge);
    final_reduce_wmma<<<1, 32, 0, stream>>>(hinge, out, T);
}